// CasTS_27212912787605
// MI455X (gfx1250) — compile-verified
//
#include <hip/hip_runtime.h>
#include <hip/hip_bf16.h>
#include <math.h>

typedef __attribute__((ext_vector_type(2))) float v2f;
typedef __attribute__((ext_vector_type(8))) float v8f;

#define NT 4   // N-tiles (16 cols each) per wave -> 16x64 output strip

__device__ inline float wave_sum(float v) {
#pragma unroll
  for (int off = 16; off > 0; off >>= 1) v += __shfl_xor(v, off, 32);
  return v;
}

// ---------------------------------------------------------------------------
// Generic fp32 WMMA GEMM: C = epilogue(A@B [+ A2@B2] + bias)
// MODE 0: store, 1: relu, 2: sigmoid, 3: fused GRU update (in-place auxH)
// A fragment: a[j] = A[m][k + kbase + j], kbase = (lane>>4)*2
// B fragment: b[j] = B[k + kbase + j][n],  n = lane&15
// C/D: acc[v] at row = v + (lane>>4)*8, col = lane&15
// ---------------------------------------------------------------------------
template <int MODE>
__global__ __launch_bounds__(256) void wmma_gemm_f32(
    const float* __restrict__ A, int lda, const float* __restrict__ B, int ldb,
    const float* __restrict__ A2, int lda2, const float* __restrict__ B2, int K2,
    const float* __restrict__ bias, float* __restrict__ C, int ldc,
    int M, int N, int K,
    const float* __restrict__ auxXC, const float* __restrict__ auxZR,
    float* __restrict__ auxH) {
  const int lane = threadIdx.x & 31;
  const int wave = blockIdx.x * (blockDim.x >> 5) + (threadIdx.x >> 5);
  const int ntn = N / (16 * NT);
  const int tm = wave / ntn;
  const int tn = wave % ntn;
  if (tm * 16 >= M) return;  // wave-uniform: EXEC stays all-ones
  const int m = tm * 16 + (lane & 15);
  const int kb = (lane >> 4) * 2;
  const int n0 = tn * (16 * NT) + (lane & 15);

  v8f acc[NT] = {};

  for (int k = 0; k < K; k += 4) {
    v2f a = *(const v2f*)(A + (size_t)m * lda + k + kb);
#pragma unroll
    for (int j = 0; j < NT; ++j) {
      v2f b;
      b[0] = B[(size_t)(k + kb) * ldb + n0 + j * 16];
      b[1] = B[(size_t)(k + kb + 1) * ldb + n0 + j * 16];
      acc[j] = __builtin_amdgcn_wmma_f32_16x16x4_f32(false, a, false, b,
                                                     (short)0, acc[j], false, false);
    }
  }
  if (A2 != nullptr) {
    for (int k = 0; k < K2; k += 4) {
      v2f a = *(const v2f*)(A2 + (size_t)m * lda2 + k + kb);
#pragma unroll
      for (int j = 0; j < NT; ++j) {
        v2f b;
        b[0] = B2[(size_t)(k + kb) * ldb + n0 + j * 16];
        b[1] = B2[(size_t)(k + kb + 1) * ldb + n0 + j * 16];
        acc[j] = __builtin_amdgcn_wmma_f32_16x16x4_f32(false, a, false, b,
                                                       (short)0, acc[j], false, false);
      }
    }
  }

#pragma unroll
  for (int j = 0; j < NT; ++j) {
    const int col = n0 + j * 16;
    const float bv = bias ? bias[col] : 0.f;
#pragma unroll
    for (int v = 0; v < 8; ++v) {
      const int row = tm * 16 + v + ((lane >> 4) << 3);
      float x = acc[j][v] + bv;
      if (MODE == 1) {
        x = fmaxf(x, 0.f);
        C[(size_t)row * ldc + col] = x;
      } else if (MODE == 2) {
        x = 1.f / (1.f + expf(-x));
        C[(size_t)row * ldc + col] = x;
      } else if (MODE == 3) {
        const float xc = auxXC[(size_t)row * 512 + col];
        const float z = auxZR[(size_t)row * 1024 + col];  // z = first 512 cols
        const float ho = auxH[(size_t)row * 512 + col];
        const float c = tanhf(x + xc);
        auxH[(size_t)row * 512 + col] = z * ho + (1.f - z) * c;
      } else {
        C[(size_t)row * ldc + col] = x;
      }
    }
  }
}

// RH[n,d] = r[n,d] * H[n,d],  r = ZR[:, 512:1024]
__global__ __launch_bounds__(256) void rh_kernel(const float* __restrict__ ZR,
                                                 const float* __restrict__ H,
                                                 float* __restrict__ RH) {
  const int i = blockIdx.x * 256 + threadIdx.x;  // 4096*512 elements
  const int n = i >> 9, d = i & 511;
  RH[i] = ZR[(size_t)n * 1024 + 512 + d] * H[i];
}

// Row L2 normalize: out[r,:] = in[r,:] / max(||in[r,:]||, 1e-12)
__global__ __launch_bounds__(256) void l2norm_rows(const float* __restrict__ in,
                                                   float* __restrict__ out,
                                                   int D, int rows) {
  const int lane = threadIdx.x & 31;
  const int row = blockIdx.x * 8 + (threadIdx.x >> 5);
  if (row >= rows) return;
  const float* r = in + (size_t)row * D;
  float ss = 0.f;
  for (int d = lane; d < D; d += 32) { float v = r[d]; ss += v * v; }
  ss = wave_sum(ss);
  const float inv = 1.f / fmaxf(sqrtf(ss), 1e-12f);
  for (int d = lane; d < D; d += 32) out[(size_t)row * D + d] = r[d] * inv;
}

// lvec[n] = sigmoid(h_trans[n]·W_leaf + b_leaf); fin[p,n] = lvec[n]*grid_x[p,n]
__global__ __launch_bounds__(256) void grid_lvec_kernel(
    const float* __restrict__ xfeat, const float* __restrict__ Wconv,
    const float* __restrict__ bconv, const float* __restrict__ htr,
    const float* __restrict__ Wleaf, const float* __restrict__ bleaf,
    float* __restrict__ lvec, float* __restrict__ fin) {
  const int lane = threadIdx.x & 31;
  const int n = blockIdx.x * 8 + (threadIdx.x >> 5);
  float s = 0.f;
  for (int d = lane; d < 512; d += 32) s += htr[(size_t)n * 512 + d] * Wleaf[d];
  s = wave_sum(s);
  const float l = 1.f / (1.f + expf(-(s + bleaf[0])));
  if (lane == 0) lvec[n] = l;
  if (lane < 4) {
    float g = bconv[lane];
    for (int tt = 0; tt < 5; ++tt) g += xfeat[tt * 4096 + n] * Wconv[tt * 4 + lane];
    fin[lane * 4096 + n] = l * g;
  }
}

// proj[t,n] = h_trans[n] · hn[t]
__global__ __launch_bounds__(256) void proj_kernel(const float* __restrict__ htr,
                                                   const float* __restrict__ hn,
                                                   float* __restrict__ proj) {
  const int lane = threadIdx.x & 31;
  const int w = blockIdx.x * 8 + (threadIdx.x >> 5);  // 0..16383
  const int t = w >> 12, n = w & 4095;
  float s = 0.f;
  for (int d = lane; d < 512; d += 32)
    s += htr[(size_t)n * 512 + d] * hn[t * 512 + d];
  s = wave_sum(s);
  if (lane == 0) proj[t * 4096 + n] = s;
}

// U[t,:] = h_transT @ fin[t];  S[t] = sum_n fin[t,n]*proj[t,n]   (deterministic)
__global__ __launch_bounds__(256) void reduce_u_kernel(
    const float* __restrict__ htr, const float* __restrict__ fin,
    const float* __restrict__ proj, float* __restrict__ U, float* __restrict__ S) {
  const int t = blockIdx.x;
  const int tid = threadIdx.x;
  float u0 = 0.f, u1 = 0.f;
  for (int n = 0; n < 4096; ++n) {
    const float f = fin[t * 4096 + n];
    u0 += f * htr[(size_t)n * 512 + tid];
    u1 += f * htr[(size_t)n * 512 + tid + 256];
  }
  U[t * 512 + tid] = u0;
  U[t * 512 + tid + 256] = u1;
  float sp = 0.f;
  for (int n = tid; n < 4096; n += 256) sp += fin[t * 4096 + n] * proj[t * 4096 + n];
  __shared__ float sh[256];
  sh[tid] = sp; __syncthreads();
  for (int k = 128; k > 0; k >>= 1) { if (tid < k) sh[tid] += sh[tid + k]; __syncthreads(); }
  if (tid == 0) S[t] = sh[0];
}

// u <- u - 2*hn*s (finalized), W[t] = hn·u
__global__ __launch_bounds__(128) void finalize_u_kernel(const float* __restrict__ hn,
                                                         float* __restrict__ U,
                                                         const float* __restrict__ S,
                                                         float* __restrict__ W) {
  const int lane = threadIdx.x & 31;
  const int t = threadIdx.x >> 5;
  if (t >= 4) return;
  const float s = S[t];
  float w = 0.f;
  for (int d = lane; d < 512; d += 32) {
    const float u = U[t * 512 + d] - 2.f * hn[t * 512 + d] * s;
    U[t * 512 + d] = u;
    w += hn[t * 512 + d] * u;
  }
  w = wave_sum(w);
  if (lane == 0) W[t] = w;
}

// att[t,n] = h_trans[n]·u[t] - 2*proj[t,n]*W[t]
__global__ __launch_bounds__(256) void att_kernel(const float* __restrict__ htr,
                                                  const float* __restrict__ U,
                                                  const float* __restrict__ proj,
                                                  const float* __restrict__ W,
                                                  float* __restrict__ out) {
  const int lane = threadIdx.x & 31;
  const int w = blockIdx.x * 8 + (threadIdx.x >> 5);
  const int t = w >> 12, n = w & 4095;
  float s = 0.f;
  for (int d = lane; d < 512; d += 32)
    s += htr[(size_t)n * 512 + d] * U[t * 512 + d];
  s = wave_sum(s);
  if (lane == 0) out[t * 4096 + n] = s - 2.f * proj[t * 4096 + n] * W[t];
}

// Triplet margin sums per (t,m): uses Householder identity + precomputed proj
__global__ __launch_bounds__(256) void loss_off_kernel(const float* __restrict__ htr,
                                                       const float* __restrict__ proj,
                                                       const int* __restrict__ temp_y,
                                                       float* __restrict__ loff) {
  const int lane = threadIdx.x & 31;
  const int w = blockIdx.x * 8 + (threadIdx.x >> 5);  // 0..8191
  const int t = w >> 11, m = w & 2047;
  const int* row = temp_y + ((size_t)t * 2048 + m) * 18;
  const int ip = row[1];
  float p[16];
#pragma unroll
  for (int i = 0; i < 16; ++i) p[i] = htr[(size_t)ip * 512 + lane + i * 32];
  float pp = 0.f;
#pragma unroll
  for (int i = 0; i < 16; ++i) pp += p[i] * p[i];
  pp = wave_sum(pp);
  const float ppos = proj[t * 4096 + ip];
  const float apos = pp - 2.f * ppos * ppos;  // a·pos
  float lsum = 0.f;
  for (int k = 0; k < 16; ++k) {
    const int in2 = row[2 + k];
    float d = 0.f;
#pragma unroll
    for (int i = 0; i < 16; ++i) d += p[i] * htr[(size_t)in2 * 512 + lane + i * 32];
    d = wave_sum(d);
    const float aneg = d - 2.f * ppos * proj[t * 4096 + in2];
    lsum += fmaxf(aneg - apos + 0.5f, 0.f);
  }
  if (lane == 0) loff[w] = lsum;
}

// BCE-style leaf loss block partials
__global__ __launch_bounds__(256) void loss_leaf_kernel(const float* __restrict__ lvec,
                                                        const float* __restrict__ is_leaf,
                                                        float* __restrict__ lleaf) {
  const int i = blockIdx.x * 256 + threadIdx.x;  // 0..16383
  const int t = i >> 12, n = i & 4095;
  const float l = lvec[n];
  const float y = is_leaf[t * 4096 + n];
  const float term = fmaxf(l, 0.f) - l * y + log1pf(expf(-fabsf(l)));
  __shared__ float sh[256];
  sh[threadIdx.x] = term; __syncthreads();
  for (int k = 128; k > 0; k >>= 1) {
    if (threadIdx.x < k) sh[threadIdx.x] += sh[threadIdx.x + k];
    __syncthreads();
  }
  if (threadIdx.x == 0) lleaf[blockIdx.x] = sh[0];
}

__global__ __launch_bounds__(256) void final_losses_kernel(const float* __restrict__ loff,
                                                           const float* __restrict__ lleaf,
                                                           float* __restrict__ out) {
  __shared__ float sh[256];
  const int tid = threadIdx.x;
  float s = 0.f;
  for (int i = tid; i < 8192; i += 256) s += loff[i];
  sh[tid] = s; __syncthreads();
  for (int k = 128; k > 0; k >>= 1) { if (tid < k) sh[tid] += sh[tid + k]; __syncthreads(); }
  if (tid == 0) out[0] = sh[0] / (4.f * 2048.f * 16.f);
  __syncthreads();
  float s2 = 0.f;
  for (int i = tid; i < 64; i += 256) s2 += lleaf[i];
  sh[tid] = s2; __syncthreads();
  for (int k = 128; k > 0; k >>= 1) { if (tid < k) sh[tid] += sh[tid + k]; __syncthreads(); }
  if (tid == 0) out[1] = sh[0] / 16384.f;
}

extern "C" void kernel_launch(void* const* d_in, const int* in_sizes, int n_in,
                              void* d_out, int out_size, void* d_ws, size_t ws_size,
                              hipStream_t stream) {
  (void)in_sizes; (void)n_in; (void)out_size; (void)ws_size;
  const float* x_feature   = (const float*)d_in[0];   // (1,5,4096,1)
  const float* x_graph     = (const float*)d_in[1];   // (1,8,4096,128)
  const float* x_graph_inv = (const float*)d_in[2];   // (1,8,4096,128)
  const int*   temp_y      = (const int*)d_in[3];     // (4,2048,18)
  const float* is_leaf     = (const float*)d_in[4];   // (4,4096)
  const float* W_gcn       = (const float*)d_in[5];   // (128,256)
  const float* b_gcn       = (const float*)d_in[6];
  const float* Wx_zr       = (const float*)d_in[7];   // (512,1024)
  const float* Wh_zr       = (const float*)d_in[8];
  const float* b_zr        = (const float*)d_in[9];
  const float* Wx_c        = (const float*)d_in[10];  // (512,512)
  const float* Wh_c        = (const float*)d_in[11];
  const float* b_c         = (const float*)d_in[12];
  const float* h_vecs      = (const float*)d_in[13];  // (4,512)
  const float* W_leaf      = (const float*)d_in[14];  // (512,1)
  const float* b_leaf      = (const float*)d_in[15];
  const float* W_conv      = (const float*)d_in[16];  // (5,4)
  const float* b_conv      = (const float*)d_in[17];
  float* out = (float*)d_out;   // att(4*4096) | loss_off | loss_leaf

  float* ws = (float*)d_ws;
  float* X    = ws;                 // 4096*512  GCN output (X_t)
  float* H    = X   + 4096 * 512;   // 4096*512  GRU state
  float* ZR   = H   + 4096 * 512;   // 4096*1024
  float* XC   = ZR  + 4096 * 1024;  // 4096*512
  float* RH   = XC  + 4096 * 512;   // 4096*512
  float* HTR  = RH  + 4096 * 512;   // 4096*512  h_trans
  float* HN   = HTR + 4096 * 512;   // 4*512
  float* PROJ = HN  + 4 * 512;      // 4*4096
  float* FIN  = PROJ + 4 * 4096;    // 4*4096
  float* LVEC = FIN  + 4 * 4096;    // 4096
  float* U    = LVEC + 4096;        // 4*512
  float* S    = U    + 4 * 512;     // 4
  float* WT   = S    + 4;           // 4
  float* LOFF = WT   + 4;           // 8192
  float* LLEAF= LOFF + 8192;        // 64

  hipMemsetAsync(H, 0, (size_t)4096 * 512 * sizeof(float), stream);  // h0 = 0

  for (int t = 0; t < 8; ++t) {
    const float* xg  = x_graph     + (size_t)t * 4096 * 128;
    const float* xgi = x_graph_inv + (size_t)t * 4096 * 128;
    // GCN: X[:, :256]=relu(xg@W+b), X[:,256:]=relu(xgi@W+b)  (M=4096,N=256,K=128)
    wmma_gemm_f32<1><<<128, 256, 0, stream>>>(xg, 128, W_gcn, 256, nullptr, 0,
                                              nullptr, 0, b_gcn, X, 512, 4096, 256,
                                              128, nullptr, nullptr, nullptr);
    wmma_gemm_f32<1><<<128, 256, 0, stream>>>(xgi, 128, W_gcn, 256, nullptr, 0,
                                              nullptr, 0, b_gcn, X + 256, 512, 4096,
                                              256, 128, nullptr, nullptr, nullptr);
    // ZR = sigmoid(X@Wx_zr + H@Wh_zr + b_zr)   (N=1024)
    wmma_gemm_f32<2><<<512, 256, 0, stream>>>(X, 512, Wx_zr, 1024, H, 512, Wh_zr,
                                              512, b_zr, ZR, 1024, 4096, 1024, 512,
                                              nullptr, nullptr, nullptr);
    // XC = X@Wx_c + b_c
    wmma_gemm_f32<0><<<256, 256, 0, stream>>>(X, 512, Wx_c, 512, nullptr, 0,
                                              nullptr, 0, b_c, XC, 512, 4096, 512,
                                              512, nullptr, nullptr, nullptr);
    // RH = r*H
    rh_kernel<<<8192, 256, 0, stream>>>(ZR, H, RH);
    // H = z*H + (1-z)*tanh(RH@Wh_c + XC)
    wmma_gemm_f32<3><<<256, 256, 0, stream>>>(RH, 512, Wh_c, 512, nullptr, 0,
                                              nullptr, 0, nullptr, nullptr, 512,
                                              4096, 512, 512, XC, ZR, H);
  }

  l2norm_rows<<<512, 256, 0, stream>>>(H, HTR, 512, 4096);      // h_trans
  l2norm_rows<<<1, 256, 0, stream>>>(h_vecs, HN, 512, 4);       // hn
  grid_lvec_kernel<<<512, 256, 0, stream>>>(x_feature, W_conv, b_conv, HTR,
                                            W_leaf, b_leaf, LVEC, FIN);
  proj_kernel<<<2048, 256, 0, stream>>>(HTR, HN, PROJ);
  reduce_u_kernel<<<4, 256, 0, stream>>>(HTR, FIN, PROJ, U, S);
  finalize_u_kernel<<<1, 128, 0, stream>>>(HN, U, S, WT);
  att_kernel<<<2048, 256, 0, stream>>>(HTR, U, PROJ, WT, out);
  loss_off_kernel<<<1024, 256, 0, stream>>>(HTR, PROJ, temp_y, LOFF);
  loss_leaf_kernel<<<64, 256, 0, stream>>>(LVEC, is_leaf, LLEAF);
  final_losses_kernel<<<1, 256, 0, stream>>>(LOFF, LLEAF, out + 4 * 4096);
}